// GATLayer_82411832476221
// MI455X (gfx1250) — compile-verified
//
#include <hip/hip_runtime.h>

typedef __attribute__((ext_vector_type(2))) float v2f;
typedef __attribute__((ext_vector_type(8))) float v8f;

#define IN_F 128
#define OUT_F 64
#define NEG_SLOPE 0.2f
#define LDS_STRIDE (IN_F + 2)   // pad to avoid LDS bank conflicts (64 banks)

// ---------------------------------------------------------------------------
// Kernel 1: h = x @ W_v + b_v  (WMMA f32 16x16x4), fused q = h@w_q+b_q,
//           k = h@w_k+b_k via cross-lane butterfly reduction.
// One wave32 per 16-row tile; 8 waves / 256-thread block.
// ---------------------------------------------------------------------------
__global__ __launch_bounds__(256) void gat_gemm_kernel(
    const float* __restrict__ x,  const float* __restrict__ Wv,
    const float* __restrict__ bv, const float* __restrict__ wq,
    const float* __restrict__ bq, const float* __restrict__ wk,
    const float* __restrict__ bk,
    float* __restrict__ h, float* __restrict__ qout, float* __restrict__ kout,
    int N)
{
    __shared__ float Wt[OUT_F * LDS_STRIDE];   // W_v transposed: Wt[n][k]

    const int tid = threadIdx.x;
    // cooperative transpose of W_v [128][64] into LDS
    for (int idx = tid; idx < IN_F * OUT_F; idx += 256) {
        int kk = idx / OUT_F, n = idx % OUT_F;
        Wt[n * LDS_STRIDE + kk] = Wv[idx];
    }
    __syncthreads();

    const int wave = tid >> 5;
    const int lane = tid & 31;
    const int half = lane >> 4;      // 0 | 1
    const int l16  = lane & 15;
    const long tile = (long)blockIdx.x * 8 + wave;
    const long rowBase = tile * 16;
    if (rowBase >= (long)N) return;  // uniform across wave: EXEC stays all-1s

    // C/D layout: VGPR j -> row m = j + 8*half ; lane -> col n = nt*16 + l16
    v8f acc[4];
#pragma unroll
    for (int nt = 0; nt < 4; ++nt) {
        float b = bv[nt * 16 + l16];
#pragma unroll
        for (int j = 0; j < 8; ++j) acc[nt][j] = b;
    }

    long arow = rowBase + l16;
    if (arow >= (long)N) arow = (long)N - 1;   // clamp (predication-free)
    const float* __restrict__ xrow = x + arow * IN_F;

    for (int kk = 0; kk < IN_F; kk += 4) {
        const int kl = kk + 2 * half;          // A/B K-index base for this lane
        const float2 av = *(const float2*)(xrow + kl);
        v2f a; a.x = av.x; a.y = av.y;
#pragma unroll
        for (int nt = 0; nt < 4; ++nt) {
            const float* wp = &Wt[(nt * 16 + l16) * LDS_STRIDE + kl];
            v2f b; b.x = wp[0]; b.y = wp[1];
            acc[nt] = __builtin_amdgcn_wmma_f32_16x16x4_f32(
                false, a, false, b, (short)0, acc[nt], false, false);
        }
    }

    // fused q/k projections: each lane owns 4 columns (n = nt*16 + l16)
    float qpart[8], kpart[8];
#pragma unroll
    for (int j = 0; j < 8; ++j) { qpart[j] = 0.f; kpart[j] = 0.f; }
#pragma unroll
    for (int nt = 0; nt < 4; ++nt) {
        const float wqn = wq[nt * 16 + l16];
        const float wkn = wk[nt * 16 + l16];
#pragma unroll
        for (int j = 0; j < 8; ++j) {
            qpart[j] += acc[nt][j] * wqn;
            kpart[j] += acc[nt][j] * wkn;
        }
    }
    // reduce over the 16 lanes of each half (masks <=8 never cross halves)
#pragma unroll
    for (int off = 8; off >= 1; off >>= 1) {
#pragma unroll
        for (int j = 0; j < 8; ++j) {
            qpart[j] += __shfl_xor(qpart[j], off, 32);
            kpart[j] += __shfl_xor(kpart[j], off, 32);
        }
    }

    // store h tile (coalesced 64B per half-wave per row)
#pragma unroll
    for (int nt = 0; nt < 4; ++nt) {
#pragma unroll
        for (int j = 0; j < 8; ++j) {
            const long m = rowBase + half * 8 + j;
            if (m < (long)N) h[m * OUT_F + nt * 16 + l16] = acc[nt][j];
        }
    }
    if (l16 == 0) {
        const float bqv = bq[0], bkv = bk[0];
#pragma unroll
        for (int j = 0; j < 8; ++j) {
            const long m = rowBase + half * 8 + j;
            if (m < (long)N) {
                qout[m] = qpart[j] + bqv;
                kout[m] = kpart[j] + bkv;
            }
        }
    }
}

// order-preserving float<->uint mapping for atomic max
__device__ __forceinline__ unsigned f2ord(float f) {
    unsigned u = __float_as_uint(f);
    return (u & 0x80000000u) ? ~u : (u | 0x80000000u);
}
__device__ __forceinline__ float ord2f(unsigned u) {
    return (u & 0x80000000u) ? __uint_as_float(u ^ 0x80000000u)
                             : __uint_as_float(~u);
}

// ---------------------------------------------------------------------------
// Kernel 2: coeff = leaky_relu(q[src]+k[dst]); segment max into mmax[dst]
// ---------------------------------------------------------------------------
__global__ __launch_bounds__(256) void edge_max_kernel(
    const float* __restrict__ q, const float* __restrict__ k,
    const int* __restrict__ src, const int* __restrict__ dst,
    float* __restrict__ coeff, unsigned* __restrict__ mmax, int E)
{
    int i = blockIdx.x * 256 + threadIdx.x;
    if (i >= E) return;
    const int s = src[i], d = dst[i];
    const float e = q[s] + k[d];
    const float c = (e >= 0.f) ? e : NEG_SLOPE * e;
    coeff[i] = c;
    atomicMax(&mmax[d], f2ord(c));
}

// ---------------------------------------------------------------------------
// Kernel 3: ex = exp(coeff - m[dst]); segment sum into ssum[dst]
// (coeff buffer is overwritten in place with ex — each thread owns its slot)
// ---------------------------------------------------------------------------
__global__ __launch_bounds__(256) void edge_expsum_kernel(
    const int* __restrict__ dst, const unsigned* __restrict__ mmax,
    float* __restrict__ coeff, float* __restrict__ ssum, int E)
{
    int i = blockIdx.x * 256 + threadIdx.x;
    if (i >= E) return;
    const int d = dst[i];
    const float m = ord2f(mmax[d]);
    const float ex = __expf(coeff[i] - m);
    coeff[i] = ex;
    atomicAdd(&ssum[d], ex);
}

// ---------------------------------------------------------------------------
// Kernel 4: out[dst] += h[src] * (ex / ssum[dst]) ; one wave32 per edge,
// 2 features per lane (float2 coalesced gather, fp32 atomic scatter to L2)
// ---------------------------------------------------------------------------
__global__ __launch_bounds__(256) void edge_aggregate_kernel(
    const float* __restrict__ h, const int* __restrict__ src,
    const int* __restrict__ dst, const float* __restrict__ ex,
    const float* __restrict__ ssum, float* __restrict__ out, int E)
{
    const long gwave = ((long)blockIdx.x * 256 + threadIdx.x) >> 5;
    const int lane = threadIdx.x & 31;
    if (gwave >= (long)E) return;
    const int s = src[gwave], d = dst[gwave];
    const float attn = ex[gwave] / ssum[d];
    const float2 v = *(const float2*)(h + (long)s * OUT_F + 2 * lane);
    float* op = out + (long)d * OUT_F + 2 * lane;
    atomicAdd(op,     v.x * attn);
    atomicAdd(op + 1, v.y * attn);
}

// ---------------------------------------------------------------------------
extern "C" void kernel_launch(void* const* d_in, const int* in_sizes, int n_in,
                              void* d_out, int out_size, void* d_ws, size_t ws_size,
                              hipStream_t stream) {
    const float* x  = (const float*)d_in[0];
    const float* Wv = (const float*)d_in[1];
    const float* bv = (const float*)d_in[2];
    const float* wq = (const float*)d_in[3];
    const float* bq = (const float*)d_in[4];
    const float* wk = (const float*)d_in[5];
    const float* bk = (const float*)d_in[6];
    const int*   src = (const int*)d_in[7];
    const int*   dst = (const int*)d_in[8];

    const int N = in_sizes[0] / IN_F;
    const int E = in_sizes[7];
    float* out = (float*)d_out;

    // carve workspace (256B aligned slices)
    char* p = (char*)d_ws;
    auto carve = [&](size_t bytes) {
        char* r = p;
        p += (bytes + 255) & ~(size_t)255;
        return r;
    };
    float*    h    = (float*)   carve((size_t)N * OUT_F * sizeof(float));
    float*    q    = (float*)   carve((size_t)N * sizeof(float));
    float*    k    = (float*)   carve((size_t)N * sizeof(float));
    unsigned* mmax = (unsigned*)carve((size_t)N * sizeof(unsigned));
    float*    ssum = (float*)   carve((size_t)N * sizeof(float));
    float*    ex   = (float*)   carve((size_t)E * sizeof(float));

    // re-initialize accumulators every call (graph-capture safe)
    hipMemsetAsync(out,  0, (size_t)N * OUT_F * sizeof(float), stream);
    hipMemsetAsync(mmax, 0, (size_t)N * sizeof(unsigned), stream);  // 0 < ord(-inf)
    hipMemsetAsync(ssum, 0, (size_t)N * sizeof(float), stream);

    const int tiles = (N + 15) / 16;
    const int gemmBlocks = (tiles + 7) / 8;
    gat_gemm_kernel<<<gemmBlocks, 256, 0, stream>>>(
        x, Wv, bv, wq, bq, wk, bk, h, q, k, N);

    edge_max_kernel<<<(E + 255) / 256, 256, 0, stream>>>(
        q, k, src, dst, ex, mmax, E);

    edge_expsum_kernel<<<(E + 255) / 256, 256, 0, stream>>>(
        dst, mmax, ex, ssum, E);

    edge_aggregate_kernel<<<(E + 7) / 8, 256, 0, stream>>>(
        h, src, dst, ex, ssum, out, E);
}